// SemiCurrSinkhornKnopp_65128884077217
// MI455X (gfx1250) — compile-verified
//
#include <hip/hip_runtime.h>
#include <math.h>

// ---- problem constants (from reference) ----
#define NROWS 16384
#define KCOLS 2048
#define KP1   2049
#define NSPLIT 64
#define ROWS_PER_SPLIT (NROWS / NSPLIT)   // 256
#define NJTILE (KCOLS / 16)               // 128
#define FI (1.0f / 1.1f)                  // gamma/(gamma+eps)
#define INV_EPS 10.0f
#define RHO_ 0.1f
#define STOPERR_ 1e-6f
#define ITERMAX 1000

typedef __attribute__((ext_vector_type(2))) float v2f;
typedef __attribute__((ext_vector_type(8))) float v8f;

// ---------------------------------------------------------------------------
// init: b0 = 1/(k+1), flag = 0
// ---------------------------------------------------------------------------
__global__ __launch_bounds__(256) void sk_init(float* __restrict__ b, int* __restrict__ flag) {
  const int t = threadIdx.x;
  for (int j = t; j < KP1; j += 256) b[j] = 1.0f / (float)KP1;
  if (t == 0) *flag = 0;
}

// ---------------------------------------------------------------------------
// Q = exp(10 * logsoftmax(P)) per row, written in place into the output buffer.
// One workgroup per row; row cached in LDS; two tree reductions (max, sumexp).
// ---------------------------------------------------------------------------
__global__ __launch_bounds__(256) void sk_softq(const float* __restrict__ P,
                                                float* __restrict__ Q) {
  __shared__ float row[KCOLS];
  __shared__ float red[256];
  const int r = blockIdx.x;
  const int t = threadIdx.x;
  const float* p = P + (size_t)r * KCOLS;

  float m = -INFINITY;
  for (int j = t; j < KCOLS; j += 256) { float v = p[j]; row[j] = v; m = fmaxf(m, v); }
  red[t] = m; __syncthreads();
  for (int s = 128; s > 0; s >>= 1) { if (t < s) red[t] = fmaxf(red[t], red[t + s]); __syncthreads(); }
  const float rowmax = red[0];
  __syncthreads();

  float se = 0.f;
  for (int j = t; j < KCOLS; j += 256) se += expf(row[j] - rowmax);
  red[t] = se; __syncthreads();
  for (int s = 128; s > 0; s >>= 1) { if (t < s) red[t] += red[t + s]; __syncthreads(); }
  const float logZ = rowmax + logf(red[0]);

  float* q = Q + (size_t)r * KCOLS;
  for (int j = t; j < KCOLS; j += 256) q[j] = expf(INV_EPS * (row[j] - logZ));
}

// ---------------------------------------------------------------------------
// row pass: a_i = (1/n) / (sum_j Q_ij b_j + b_slack). One wave per row,
// coalesced float4 loads of Q (L2-resident), b staged in LDS, wave32 butterfly.
// ---------------------------------------------------------------------------
__global__ __launch_bounds__(256) void sk_rowpass(const float* __restrict__ Q,
                                                  const float* __restrict__ b,
                                                  float* __restrict__ a,
                                                  const int* __restrict__ flag) {
  if (*flag) return;
  __shared__ float bsh[KP1];
  const int t = threadIdx.x;
  for (int j = t; j < KP1; j += 256) bsh[j] = b[j];
  __syncthreads();

  const int wave = t >> 5, lane = t & 31;
  const int r = blockIdx.x * 8 + wave;
  const float* q = Q + (size_t)r * KCOLS;

  float acc = 0.f;
  for (int c = lane * 4; c < KCOLS; c += 128) {
    const float4 qv = *reinterpret_cast<const float4*>(q + c);
    acc = fmaf(qv.x, bsh[c + 0], acc);
    acc = fmaf(qv.y, bsh[c + 1], acc);
    acc = fmaf(qv.z, bsh[c + 2], acc);
    acc = fmaf(qv.w, bsh[c + 3], acc);
  }
  for (int off = 16; off > 0; off >>= 1) acc += __shfl_xor(acc, off, 32);
  if (lane == 0) a[r] = (1.0f / (float)NROWS) / (acc + bsh[KCOLS]);
}

// ---------------------------------------------------------------------------
// column pass: partial colsum_j = sum_i a_i * Q_ij via v_wmma_f32_16x16x4_f32.
//   A (16x4): a[r..r+3] broadcast across M  -> lane-half h, vgpr g holds a[r+2h+g]
//   B (4x16): Q[r+2h+g, jbase+n]            (64B coalesced per lane-half)
//   D (16x16): every M-row identical == 16 column partial sums; extract VGPR0.
// One wave per (16-column tile, 256-row split); partials written, no atomics.
// ---------------------------------------------------------------------------
__global__ __launch_bounds__(256) void sk_colpass(const float* __restrict__ Q,
                                                  const float* __restrict__ a,
                                                  float* __restrict__ part,
                                                  const int* __restrict__ flag) {
  if (*flag) return;
  const int t = threadIdx.x;
  const int wave = t >> 5, lane = t & 31;
  const int task = blockIdx.x * 8 + wave;          // [0, NSPLIT*NJTILE)
  const int jtile = task & (NJTILE - 1);
  const int split = task >> 7;                     // task / NJTILE
  const int jbase = jtile * 16;
  const int rbase = split * ROWS_PER_SPLIT;
  const int h2 = (lane >> 4) * 2;                  // 0 or 2
  const int n = lane & 15;

  v8f c = {0.f, 0.f, 0.f, 0.f, 0.f, 0.f, 0.f, 0.f};
  for (int r = rbase; r < rbase + ROWS_PER_SPLIT; r += 4) {
    const int qr = r + h2;
    v2f av; av.x = a[qr];     av.y = a[qr + 1];
    v2f bv;
    bv.x = Q[(size_t)qr       * KCOLS + jbase + n];
    bv.y = Q[(size_t)(qr + 1) * KCOLS + jbase + n];
    c = __builtin_amdgcn_wmma_f32_16x16x4_f32(false, av, false, bv, (short)0, c,
                                              false, false);
  }
  if ((lane >> 4) == 0) part[(size_t)split * KCOLS + jbase + n] = c[0];
}

// ---------------------------------------------------------------------------
// update: reduce partials -> b_new = (Pb/colsum)^fi ; slack = (1-rho)/sum(a);
// err = ||b_new - b||; set convergence flag. Single workgroup (tiny work).
// ---------------------------------------------------------------------------
__global__ __launch_bounds__(256) void sk_update(const float* __restrict__ part,
                                                 const float* __restrict__ a,
                                                 float* __restrict__ b,
                                                 int* __restrict__ flag) {
  if (*flag) return;
  __shared__ float red[256];
  const int t = threadIdx.x;

  float sa = 0.f;
  for (int i = t; i < NROWS; i += 256) sa += a[i];
  red[t] = sa; __syncthreads();
  for (int s = 128; s > 0; s >>= 1) { if (t < s) red[t] += red[t + s]; __syncthreads(); }
  const float sum_a = red[0];
  __syncthreads();

  float e2 = 0.f;
  const float Pb = RHO_ / (float)KCOLS;
  for (int j = t; j < KCOLS; j += 256) {
    float cs = 0.f;
    for (int s = 0; s < NSPLIT; ++s) cs += part[(size_t)s * KCOLS + j];
    const float bn = powf(Pb / cs, FI);
    const float d = bn - b[j];
    e2 += d * d;
    b[j] = bn;
  }
  if (t == 0) {
    const float bn = (1.0f - RHO_) / sum_a;   // slack column: no pow
    const float d = bn - b[KCOLS];
    e2 += d * d;
    b[KCOLS] = bn;
  }
  red[t] = e2; __syncthreads();
  for (int s = 128; s > 0; s >>= 1) { if (t < s) red[t] += red[t + s]; __syncthreads(); }
  if (t == 0 && sqrtf(red[0]) <= STOPERR_) *flag = 1;
}

// ---------------------------------------------------------------------------
// finalize (in place): out_ij = n * a_i * Q_ij * b_j   (slack column dropped)
// ---------------------------------------------------------------------------
__global__ __launch_bounds__(256) void sk_finalize(float* __restrict__ out,
                                                   const float* __restrict__ a,
                                                   const float* __restrict__ b) {
  const size_t idx4 = ((size_t)blockIdx.x * 256 + threadIdx.x) * 4;
  const int i = (int)(idx4 >> 11);      // / 2048
  const int j = (int)(idx4 & 2047);
  float4 q = *reinterpret_cast<float4*>(out + idx4);
  const float s = (float)NROWS * a[i];
  q.x *= s * b[j + 0];
  q.y *= s * b[j + 1];
  q.z *= s * b[j + 2];
  q.w *= s * b[j + 3];
  *reinterpret_cast<float4*>(out + idx4) = q;
}

// ---------------------------------------------------------------------------
extern "C" void kernel_launch(void* const* d_in, const int* in_sizes, int n_in,
                              void* d_out, int out_size, void* d_ws, size_t ws_size,
                              hipStream_t stream) {
  (void)in_sizes; (void)n_in; (void)out_size; (void)ws_size;
  const float* P = (const float*)d_in[0];
  float* Q = (float*)d_out;                 // Q built in place in the output buffer

  float* W    = (float*)d_ws;
  float* a    = W;                          // 16384 floats
  float* b    = W + NROWS;                  // 2049 floats (4096 reserved)
  float* part = W + NROWS + 4096;           // NSPLIT*2048 floats
  int*   flag = (int*)(part + (size_t)NSPLIT * KCOLS);

  sk_init<<<1, 256, 0, stream>>>(b, flag);
  sk_softq<<<NROWS, 256, 0, stream>>>(P, Q);
  for (int it = 0; it < ITERMAX; ++it) {
    sk_rowpass<<<NROWS / 8, 256, 0, stream>>>(Q, b, a, flag);
    sk_colpass<<<(NSPLIT * NJTILE) / 8, 256, 0, stream>>>(Q, a, part, flag);
    sk_update<<<1, 256, 0, stream>>>(part, a, b, flag);
  }
  sk_finalize<<<(NROWS * KCOLS) / 1024, 256, 0, stream>>>(Q, a, b);
}